// MVN3D_26207890440796
// MI455X (gfx1250) — compile-verified
//
#include <hip/hip_runtime.h>
#include <math.h>

#define BATCH   1323
#define NPTS    1323
#define DMODEL  1024
#define LOG2PI  1.8378770664093453f   // log(2*pi)

typedef __attribute__((ext_vector_type(2))) float v2f;
typedef __attribute__((ext_vector_type(8))) float v8f;

// ---------------------------------------------------------------------------
// Kernel 1: proj[b, 0..8] = rep[b,:] @ Wcat[0..8,:]^T   (Wcat = [Wm;Ws] rows)
// One wave32 per 16-row tile of rep. WMMA f32 16x16x4, K looped over 1024.
//   A (16x4 f32): lanes 0-15 hold M=lane, v0=K0,v1=K1; lanes 16-31: K2,K3.
//   B (4x16 f32): lanes 0-15 hold N=lane, v0=K0,v1=K1; lanes 16-31: K2,K3.
//   C (16x16 f32): VGPR r = row r (lanes 0-15) / row r+8 (lanes 16-31), N=lane&15.
// ---------------------------------------------------------------------------
__global__ void __launch_bounds__(32)
mvn_proj_wmma(const float* __restrict__ rep,
              const float* __restrict__ Wm,
              const float* __restrict__ Ws,
              float* __restrict__ proj /* BATCH x 9 */) {
    const int lane = threadIdx.x;          // 0..31
    const int half = lane >> 4;            // 0 | 1
    const int l    = lane & 15;

    // A-row (clamped so out-of-range rows read valid memory; they are never stored)
    int row = blockIdx.x * 16 + l;
    int rowc = row < BATCH ? row : (BATCH - 1);
    const float* arow = rep + (size_t)rowc * DMODEL;

    // B-column n = l -> row n of Wcat. n in [9,16) clamped to a valid Ws row;
    // garbage stays in columns 9..15 which are never stored.
    const float* wrow;
    if (l < 3)       wrow = Wm + (size_t)l * DMODEL;
    else if (l < 9)  wrow = Ws + (size_t)(l - 3) * DMODEL;
    else             wrow = Ws;   // dummy valid row

    const int ko = half * 2;   // this half-wave supplies K = k+ko, k+ko+1

    v8f acc = {0.f, 0.f, 0.f, 0.f, 0.f, 0.f, 0.f, 0.f};
#pragma unroll 4
    for (int k = 0; k < DMODEL; k += 4) {
        v2f a = *(const v2f*)(arow + k + ko);
        v2f b = *(const v2f*)(wrow + k + ko);
        // 8 args: (neg_a, A, neg_b, B, c_mod, C, reuse_a, reuse_b)
        acc = __builtin_amdgcn_wmma_f32_16x16x4_f32(
            false, a, false, b, (short)0, acc, false, false);
    }

    // Store columns 0..8 only.
    if (l < 9) {
#pragma unroll
        for (int r = 0; r < 8; ++r) {
            int rr = blockIdx.x * 16 + half * 8 + r;
            if (rr < BATCH) proj[(size_t)rr * 9 + l] = acc[r];
        }
    }
}

// ---------------------------------------------------------------------------
// Kernel 2: per-batch parameters + L output.
// params[b*12 + {0..9}] = m0,m1,m2, 1/L00, L10, 1/L11, L20, L21, 1/L22,
//                         (1.5*log2pi + log L00 + log L11 + log L22)
// Lout[b*9 + ..] = {L00,0,0, L10,L11,0, L20,L21,L22}
// ---------------------------------------------------------------------------
__device__ __forceinline__ float softplus_f(float x) {
    // log(1 + exp(x)) numerically stable
    return fmaxf(x, 0.f) + log1pf(__expf(-fabsf(x)));
}

__global__ void __launch_bounds__(256)
mvn_params(const float* __restrict__ proj,
           const float* __restrict__ bm,
           const float* __restrict__ bs,
           float* __restrict__ params,
           float* __restrict__ Lout) {
    int b = blockIdx.x * blockDim.x + threadIdx.x;
    if (b >= BATCH) return;
    const float* pr = proj + (size_t)b * 9;

    float m0 = pr[0] + bm[0];
    float m1 = pr[1] + bm[1];
    float m2 = pr[2] + bm[2];
    float L00 = softplus_f(pr[3] + bs[0]);
    float L10 = pr[4] + bs[1];
    float L11 = softplus_f(pr[5] + bs[2]);
    float L20 = pr[6] + bs[3];
    float L21 = pr[7] + bs[4];
    float L22 = softplus_f(pr[8] + bs[5]);

    float i00 = 1.f / L00, i11 = 1.f / L11, i22 = 1.f / L22;
    float c = 1.5f * LOG2PI + __logf(L00) + __logf(L11) + __logf(L22);

    float* pp = params + (size_t)b * 12;
    pp[0] = m0;  pp[1] = m1;  pp[2] = m2;
    pp[3] = i00; pp[4] = L10; pp[5] = i11;
    pp[6] = L20; pp[7] = L21; pp[8] = i22;
    pp[9] = c;   pp[10] = 0.f; pp[11] = 0.f;

    float* Lo = Lout + (size_t)b * 9;
    Lo[0] = L00; Lo[1] = 0.f; Lo[2] = 0.f;
    Lo[3] = L10; Lo[4] = L11; Lo[5] = 0.f;
    Lo[6] = L20; Lo[7] = L21; Lo[8] = L22;
}

// ---------------------------------------------------------------------------
// Kernel 3: the bandwidth pass. grid = (ceil(NPTS/256), BATCH).
// profile[b*NPTS + p] = exp(-0.5*M - c_b)
// ---------------------------------------------------------------------------
__global__ void __launch_bounds__(256)
mvn_profile(const float* __restrict__ dxyz,
            const float* __restrict__ params,
            float* __restrict__ profile) {
    __shared__ float P[12];
    const int b = blockIdx.y;
    if (threadIdx.x < 12) P[threadIdx.x] = params[(size_t)b * 12 + threadIdx.x];
    __syncthreads();

    int p = blockIdx.x * blockDim.x + threadIdx.x;
    if (p >= NPTS) return;

    size_t idx = (size_t)b * NPTS + p;
    const float* q = dxyz + idx * 3;   // 12 contiguous bytes -> b96 load
    float x = q[0], y = q[1], z = q[2];

    float d0 = x - P[0];
    float d1 = y - P[1];
    float d2 = z - P[2];
    float y0 = d0 * P[3];                               // /L00
    float y1 = (d1 - P[4] * y0) * P[5];                 // /L11
    float y2 = (d2 - P[6] * y0 - P[7] * y1) * P[8];     // /L22
    float M  = fmaf(y0, y0, fmaf(y1, y1, y2 * y2));
    profile[idx] = __expf(fmaf(-0.5f, M, -P[9]));
}

// ---------------------------------------------------------------------------
extern "C" void kernel_launch(void* const* d_in, const int* in_sizes, int n_in,
                              void* d_out, int out_size, void* d_ws, size_t ws_size,
                              hipStream_t stream) {
    const float* rep  = (const float*)d_in[0];
    const float* dxyz = (const float*)d_in[1];
    const float* Wm   = (const float*)d_in[2];
    const float* bm   = (const float*)d_in[3];
    const float* Ws   = (const float*)d_in[4];
    const float* bs   = (const float*)d_in[5];
    (void)in_sizes; (void)n_in; (void)out_size; (void)ws_size;

    float* out     = (float*)d_out;
    float* profile = out;                               // BATCH*NPTS floats
    float* Lout    = out + (size_t)BATCH * NPTS;        // BATCH*9 floats

    float* proj   = (float*)d_ws;                       // BATCH*9  floats
    float* params = proj + 12288;                       // BATCH*12 floats (48KB offset)

    mvn_proj_wmma<<<dim3((BATCH + 15) / 16), dim3(32), 0, stream>>>(rep, Wm, Ws, proj);
    mvn_params<<<dim3((BATCH + 255) / 256), dim3(256), 0, stream>>>(proj, bm, bs, params, Lout);
    mvn_profile<<<dim3((NPTS + 255) / 256, BATCH), dim3(256), 0, stream>>>(dxyz, params, profile);
}